// HardNegativeCELoss_74758200754290
// MI455X (gfx1250) — compile-verified
//
#include <hip/hip_runtime.h>
#include <hip/hip_bf16.h>

// ---------------- problem constants (from reference) ----------------
#define BATCH   8
#define CH      512
#define TT      1500
#define NTOK    (BATCH * TT)      // 12000
#define VOCAB   4096
#define KTOP    100

// LDS layout (floats)
#define DSTRIDE 4100              // 4096 + 4 pad -> conflict-free epilogue stores
#define ASTRIDE 516               // 512 + 4 pad  -> conflict-free ds_load_b64 A reads
#define CCAP    448               // per-token candidate buffer (14 per lane)
#define NCAND   14                // CCAP / 32
#define SMEM_FLOATS (16 * DSTRIDE + 16 * ASTRIDE + 16 + 16 * CCAP)
#define SMEM_BYTES  (SMEM_FLOATS * 4)   // 324,160 B <= 320KB WGP LDS

typedef float v2f __attribute__((ext_vector_type(2)));
typedef float v8f __attribute__((ext_vector_type(8)));

// ---------------- kernel 0: codebook squared norms -> ws ----------------
__global__ void cbsq_kernel(const float* __restrict__ cb, float* __restrict__ cbsq) {
    const int wave = threadIdx.x >> 5;          // 8 waves / block
    const int lane = threadIdx.x & 31;
    const int code = blockIdx.x * 8 + wave;     // grid = 512 blocks
    if (code >= VOCAB) return;
    const float* row = cb + (size_t)code * CH;
    float s = 0.f;
#pragma unroll
    for (int j = 0; j < CH / 32; ++j) {
        float v = row[lane + 32 * j];
        s = fmaf(v, v, s);
    }
#pragma unroll
    for (int off = 16; off > 0; off >>= 1) s += __shfl_xor(s, off, 32);
    if (lane == 0) cbsq[code] = s;
}

// ---------------- main fused kernel ----------------
// grid = NTOK/16 = 750 WGs, block = 512 threads (16 wave32s), dyn LDS = SMEM_BYTES
__global__ void __launch_bounds__(512, 1)
hardneg_ce_kernel(const float* __restrict__ emb,        // (8, 512, 1500)
                  const long long* __restrict__ codes,  // (8, 1500) int64
                  const float* __restrict__ cb,         // (4096, 512)
                  const float* __restrict__ cbsq,       // (4096,)
                  float* __restrict__ out)              // 4 floats
{
    extern __shared__ float smem[];
    float* sDist  = smem;                     // 16 * DSTRIDE
    float* sA     = smem + 16 * DSTRIDE;      // 16 * ASTRIDE
    float* sEmbSq = sA + 16 * ASTRIDE;        // 16
    float* sCand  = sEmbSq + 16;              // 16 * CCAP

    const int tid  = threadIdx.x;
    const int wave = tid >> 5;                // 0..15
    const int lane = tid & 31;
    const int tok0 = blockIdx.x * 16;

    // ---- stage A tile: 16 tokens x 512 channels, coalesced over t ----
#pragma unroll
    for (int j = 0; j < 16; ++j) {
        const int c = j * 32 + (tid >> 4);    // 0..511
        const int m = tid & 15;
        const int n = tok0 + m;
        const int b = n / TT;
        const int t = n - b * TT;
        sA[m * ASTRIDE + c] = emb[((size_t)(b * CH + c)) * TT + t];
    }
    __syncthreads();

    // ---- emb squared norms: wave w reduces token w ----
    {
        float s = 0.f;
#pragma unroll
        for (int j = 0; j < CH / 32; ++j) {
            float v = sA[wave * ASTRIDE + lane + 32 * j];
            s = fmaf(v, v, s);
        }
#pragma unroll
        for (int off = 16; off > 0; off >>= 1) s += __shfl_xor(s, off, 32);
        if (lane == 0) sEmbSq[wave] = s;
    }
    __syncthreads();

    // preload the 8 emb_sq values this lane's C-tile rows need (once, to regs)
    float eS[8];
#pragma unroll
    for (int r = 0; r < 8; ++r) eS[r] = sEmbSq[r + ((lane >> 4) << 3)];

    // ---- GEMM: each wave does 16 chunks of a 16x16 tile, K=512 via f32 WMMA ----
    const int arow = (lane & 15) * ASTRIDE + ((lane >> 4) << 1);
    for (int chunk = 0; chunk < VOCAB / 256; ++chunk) {
        const int code_base = chunk * 256 + wave * 16;
        const float cbsq_l = cbsq[code_base + (lane & 15)];
        const float* brow = cb + (size_t)(code_base + (lane & 15)) * CH + ((lane >> 4) << 1);

        v8f acc = {};
#pragma unroll 8
        for (int k0 = 0; k0 < CH; k0 += 4) {
            const float2 av = *reinterpret_cast<const float2*>(&sA[arow + k0]);  // ds_load_b64
            const float2 bv = *reinterpret_cast<const float2*>(&brow[k0]);       // global_load_b64 (L2)
            v2f a; a[0] = av.x; a[1] = av.y;
            v2f b; b[0] = bv.x; b[1] = bv.y;
            acc = __builtin_amdgcn_wmma_f32_16x16x4_f32(false, a, false, b,
                                                        (short)0, acc, false, false);
        }
#pragma unroll
        for (int r = 0; r < 8; ++r) {
            const int t = r + ((lane >> 4) << 3);
            float d2 = eS[r] + cbsq_l - 2.0f * acc[r];
            d2 = fmaxf(d2, 0.0f);
            sDist[t * DSTRIDE + code_base + (lane & 15)] = d2;
        }
    }
    __syncthreads();

    // ---- selection + loss: wave w owns token w ----
    const float* row = sDist + wave * DSTRIDE;
    float* cand = sCand + wave * CCAP;
    const int n_global = tok0 + wave;
    const int code = (int)codes[n_global];

    // pass 1: argmin (first-index tie rule) + per-lane 4 smallest (for bound tau)
    float bv_ = 3.4e38f; int bi_ = 0x7FFFFFFF;
    float s0 = 3.4e38f, s1 = 3.4e38f, s2 = 3.4e38f, s3 = 3.4e38f;
#pragma unroll 4
    for (int j = 0; j < VOCAB / 128; ++j) {
        const float4 q = reinterpret_cast<const float4*>(row)[lane + 32 * j];
        const int base = (lane + 32 * j) * 4;
        const float qv[4] = {q.x, q.y, q.z, q.w};
#pragma unroll
        for (int c = 0; c < 4; ++c) {
            const float v = qv[c];
            if (v < bv_ || (v == bv_ && base + c < bi_)) { bv_ = v; bi_ = base + c; }
            if (v < s3) {
                if (v < s2) { s3 = s2;
                    if (v < s1) { s2 = s1;
                        if (v < s0) { s1 = s0; s0 = v; } else s1 = v;
                    } else s2 = v;
                } else s3 = v;
            }
        }
    }
#pragma unroll
    for (int off = 16; off > 0; off >>= 1) {
        const float ov = __shfl_xor(bv_, off, 32);
        const int   oi = __shfl_xor(bi_, off, 32);
        if (ov < bv_ || (ov == bv_ && oi < bi_)) { bv_ = ov; bi_ = oi; }
        s3 = fmaxf(s3, __shfl_xor(s3, off, 32));
    }
    const float tau = s3;                 // >=128 elements <= tau  =>  K-th smallest <= tau
    const float d_min = sqrtf(bv_);
    const float d2_true = row[code];
    const float d_true = sqrtf(d2_true);

    // pass 2: ballot-compact all elements <= tau into per-token LDS buffer
    int cbase = 0;                        // uniform across wave
#pragma unroll 4
    for (int j = 0; j < VOCAB / 128; ++j) {
        const float4 q = reinterpret_cast<const float4*>(row)[lane + 32 * j];
        const float qv[4] = {q.x, q.y, q.z, q.w};
#pragma unroll
        for (int c = 0; c < 4; ++c) {
            const bool p = (qv[c] <= tau);
            const unsigned ball = (unsigned)__ballot(p);
            if (p) {
                const int slot = cbase + __popc(ball & ((1u << lane) - 1u));
                if (slot < CCAP) cand[slot] = qv[c];
            }
            cbase += __popc(ball);
        }
    }
    const int C = cbase;                  // total candidates (>= 128 >= KTOP)

    unsigned T;                           // bits of K-th smallest d2
    int c_lt; float S;

    if (C <= CCAP) {
        // register-resident candidates: search costs zero memory traffic
        unsigned u[NCAND];
#pragma unroll
        for (int r = 0; r < NCAND; ++r) {
            const int idx = lane + 32 * r;
            u[r] = (idx < C) ? __float_as_uint(cand[idx]) : 0x7F800000u;
        }
        unsigned lo = __float_as_uint(bv_), hi = __float_as_uint(tau);
        while (lo < hi) {
            const unsigned mid = lo + ((hi - lo) >> 1);
            int cnt = 0;
#pragma unroll
            for (int r = 0; r < NCAND; ++r) cnt += (u[r] <= mid);
#pragma unroll
            for (int off = 16; off > 0; off >>= 1) cnt += __shfl_xor(cnt, off, 32);
            if (cnt >= KTOP) hi = mid; else lo = mid + 1;
        }
        T = lo;
        c_lt = 0; S = 0.f;
#pragma unroll
        for (int r = 0; r < NCAND; ++r) {
            if (u[r] < T) { c_lt++; S += __expf(d_min - sqrtf(__uint_as_float(u[r]))); }
        }
#pragma unroll
        for (int off = 16; off > 0; off >>= 1) {
            c_lt += __shfl_xor(c_lt, off, 32);
            S    += __shfl_xor(S, off, 32);
        }
    } else {
        // fallback (statistically never): full-row bitwise binary search
        unsigned lo = __float_as_uint(bv_), hi = __float_as_uint(tau);
        while (lo < hi) {
            const unsigned mid = lo + ((hi - lo) >> 1);
            int cnt = 0;
#pragma unroll 4
            for (int j = 0; j < VOCAB / 128; ++j) {
                const float4 q = reinterpret_cast<const float4*>(row)[lane + 32 * j];
                cnt += (__float_as_uint(q.x) <= mid) + (__float_as_uint(q.y) <= mid)
                     + (__float_as_uint(q.z) <= mid) + (__float_as_uint(q.w) <= mid);
            }
#pragma unroll
            for (int off = 16; off > 0; off >>= 1) cnt += __shfl_xor(cnt, off, 32);
            if (cnt >= KTOP) hi = mid; else lo = mid + 1;
        }
        T = lo;
        c_lt = 0; S = 0.f;
#pragma unroll 4
        for (int j = 0; j < VOCAB / 128; ++j) {
            const float4 q = reinterpret_cast<const float4*>(row)[lane + 32 * j];
            const float qv[4] = {q.x, q.y, q.z, q.w};
#pragma unroll
            for (int c = 0; c < 4; ++c) {
                if (__float_as_uint(qv[c]) < T) { c_lt++; S += __expf(d_min - sqrtf(qv[c])); }
            }
        }
#pragma unroll
        for (int off = 16; off > 0; off >>= 1) {
            c_lt += __shfl_xor(c_lt, off, 32);
            S    += __shfl_xor(S, off, 32);
        }
    }

    const float d_K = sqrtf(__uint_as_float(T));
    S += (float)(KTOP - c_lt) * __expf(d_min - d_K);   // ties at T fill remaining slots

    const bool in_topk = (__float_as_uint(d2_true) <= T);
    float denom = S;
    if (!in_topk) denom = S - __expf(d_min - d_K) + __expf(d_min - d_true);

    const float loss_tok = (d_true - d_min) + __logf(denom);
    const float correct  = (bi_ == code) ? 1.0f : 0.0f;   // local == global here

    if (lane == 0) {
        const float invN = 1.0f / (float)NTOK;
        atomicAdd(&out[0], loss_tok * invN);
        atomicAdd(&out[1], correct * invN);   // local_accuracy
        atomicAdd(&out[2], correct * invN);   // global_accuracy
        atomicAdd(&out[3], invN);             // correct_in_candidates == 1 by construction
    }
}

// ---------------- host launcher ----------------
extern "C" void kernel_launch(void* const* d_in, const int* in_sizes, int n_in,
                              void* d_out, int out_size, void* d_ws, size_t ws_size,
                              hipStream_t stream) {
    const float*     emb   = (const float*)d_in[0];      // (8,512,1500) f32
    const long long* codes = (const long long*)d_in[1];  // (8,1500) i64
    const float*     cb    = (const float*)d_in[2];      // (4096,512) f32
    float* out  = (float*)d_out;                         // 4 f32 scalars
    float* cbsq = (float*)d_ws;                          // 4096 f32 scratch

    (void)in_sizes; (void)n_in; (void)out_size; (void)ws_size;

    hipFuncSetAttribute((const void*)hardneg_ce_kernel,
                        hipFuncAttributeMaxDynamicSharedMemorySize, SMEM_BYTES);

    hipMemsetAsync(out, 0, 4 * sizeof(float), stream);
    cbsq_kernel<<<dim3(VOCAB / 8), dim3(256), 0, stream>>>(cb, cbsq);
    hardneg_ce_kernel<<<dim3(NTOK / 16), dim3(512), SMEM_BYTES, stream>>>(
        emb, codes, cb, cbsq, out);
}